// EvolveGCN_43267500540703
// MI455X (gfx1250) — compile-verified
//
#include <hip/hip_runtime.h>

// ---------------------------------------------------------------------------
// EvolveGCN (eval): 2-layer GCN with symmetric normalization + self loops,
// LayerNorm + ReLU per layer.  N=50000, E=600000, F_IN=HID=128, fp32.
// GEMMs use native fp32 WMMA (v_wmma_f32_16x16x4_f32) on gfx1250.
// Edge aggregation uses native global_atomic_add_f32 (no-return form,
// STOREcnt-tracked) via inline asm so it can never fall back to a CAS loop.
// ---------------------------------------------------------------------------

#define N_FEAT 128
#define LN_EPS 1e-5f

typedef float v2f __attribute__((ext_vector_type(2)));
typedef float v8f __attribute__((ext_vector_type(8)));

// Fire-and-forget fp32 atomic add (GLOBAL_ATOMIC_ADD_F32, no return value).
__device__ __forceinline__ void atomic_add_f32(float* p, float v) {
  asm volatile("global_atomic_add_f32 %0, %1, off" : : "v"(p), "v"(v) : "memory");
}

// ---------------------------------------------------------------------------
// Init: dinv = 1.0 (self loop), zero agg buffer and d_out
// ---------------------------------------------------------------------------
__global__ void init_kernel(float* __restrict__ dinv, float* __restrict__ aggB,
                            float* __restrict__ out, int n_nodes) {
  int i = blockIdx.x * blockDim.x + threadIdx.x;
  int total = n_nodes * N_FEAT;
  if (i < total) {
    aggB[i] = 0.0f;
    out[i] = 0.0f;
  }
  if (i < n_nodes) dinv[i] = 1.0f;
}

// deg[dst] += 1 per edge (deg buffer pre-set to 1.0 for the self loop)
__global__ void deg_accum_kernel(const int* __restrict__ dst, float* __restrict__ deg,
                                 int n_edges) {
  int e = blockIdx.x * blockDim.x + threadIdx.x;
  if (e < n_edges) atomic_add_f32(&deg[dst[e]], 1.0f);
}

__global__ void deg_finalize_kernel(float* __restrict__ deg, int n_nodes) {
  int i = blockIdx.x * blockDim.x + threadIdx.x;
  if (i < n_nodes) deg[i] = rsqrtf(deg[i]);
}

// ---------------------------------------------------------------------------
// H = X @ W  (M x 128 @ 128 x 128), fp32 WMMA.
// One wave computes a 16x128 strip: 8 C tiles of 16x16, K-loop in steps of 4.
// A-frag layout (ISA 7.12.2, 32-bit A 16x4): lane -> m = lane&15,
//   kb = 2*(lane>>4); holds A[m][kb], A[m][kb+1]  (contiguous -> float2 load)
// B-frag (4x16): lane -> n = lane&15, kb = 2*(lane>>4); holds B[kb][n], B[kb+1][n]
// C/D (16x16 f32, 8 VGPRs): element r = row (r + 8*(lane>>4)), col lane&15
// M must be a multiple of 16 (50000 = 3125*16). Strip guard is wave-uniform,
// so EXEC is all-1s inside the WMMA region as required.
// ---------------------------------------------------------------------------
__global__ __launch_bounds__(256)
void gemm_wmma_f32_kernel(const float* __restrict__ X, const float* __restrict__ W,
                          float* __restrict__ H, int M) {
  const int lane = threadIdx.x & 31;
  const int strip = blockIdx.x * 8 + (threadIdx.x >> 5);
  if (strip * 16 >= M) return;   // wave-uniform

  const int row0 = strip * 16;
  const int m  = lane & 15;
  const int n  = lane & 15;
  const int kb = (lane >> 4) * 2;

  v8f acc[8] = {};   // zero accumulators

  const float* xrow = X + (size_t)(row0 + m) * N_FEAT + kb;

  for (int k0 = 0; k0 < N_FEAT; k0 += 4) {
    v2f a = *(const v2f*)(xrow + k0);                 // A[m][k0+kb], A[m][k0+kb+1]
    const float* wr0 = W + (size_t)(k0 + kb) * N_FEAT + n;
#pragma unroll
    for (int t = 0; t < 8; ++t) {
      v2f b;
      b.x = wr0[t * 16];                              // B[k0+kb  ][16t+n]
      b.y = wr0[t * 16 + N_FEAT];                     // B[k0+kb+1][16t+n]
      acc[t] = __builtin_amdgcn_wmma_f32_16x16x4_f32(
          /*neg_a=*/false, a, /*neg_b=*/false, b,
          /*c_mod=*/(short)0, acc[t], /*reuse_a=*/false, /*reuse_b=*/false);
    }
  }

  const int mhi = (lane >> 4) * 8;
#pragma unroll
  for (int t = 0; t < 8; ++t) {
    float* orow = H + (size_t)row0 * N_FEAT + t * 16 + n;
#pragma unroll
    for (int r = 0; r < 8; ++r) {
      orow[(size_t)(r + mhi) * N_FEAT] = acc[t][r];
    }
  }
}

// ---------------------------------------------------------------------------
// Edge scatter: one wave per edge.  agg[dst] += h[src] * (dinv[src]*dinv[dst])
// 128 floats = 32 lanes x float4; gather coalesced within the row,
// 4 native fp32 atomics per lane for the scatter-add.
// ---------------------------------------------------------------------------
__global__ __launch_bounds__(256)
void scatter_kernel(const int* __restrict__ src, const int* __restrict__ dst,
                    const float* __restrict__ dinv, const float* __restrict__ h,
                    float* __restrict__ agg, int n_edges) {
  int gid = blockIdx.x * blockDim.x + threadIdx.x;
  int e = gid >> 5;
  int lane = gid & 31;
  if (e >= n_edges) return;
  int s = src[e], d = dst[e];
  float w = dinv[s] * dinv[d];
  float4 hv = ((const float4*)(h + (size_t)s * N_FEAT))[lane];
  float* ap = agg + (size_t)d * N_FEAT + lane * 4;
  atomic_add_f32(ap + 0, hv.x * w);
  atomic_add_f32(ap + 1, hv.y * w);
  atomic_add_f32(ap + 2, hv.z * w);
  atomic_add_f32(ap + 3, hv.w * w);
}

// ---------------------------------------------------------------------------
// Combine + LayerNorm + ReLU: one wave per node.
// v = agg + h*selfnorm + bias; LN over 128 feats; y = relu(v_hat*g + be)
// Wave32 shuffle reductions for mean / variance.
// ---------------------------------------------------------------------------
__global__ __launch_bounds__(256)
void combine_ln_relu_kernel(const float* __restrict__ agg, const float* __restrict__ h,
                            const float* __restrict__ dinv,
                            const float* __restrict__ bias,
                            const float* __restrict__ gamma,
                            const float* __restrict__ beta,
                            float* __restrict__ out, int n_nodes) {
  int gid = blockIdx.x * blockDim.x + threadIdx.x;
  int node = gid >> 5;
  int lane = gid & 31;
  if (node >= n_nodes) return;

  float sn = dinv[node];
  sn *= sn;

  float4 a  = ((const float4*)(agg + (size_t)node * N_FEAT))[lane];
  float4 hv = ((const float4*)(h   + (size_t)node * N_FEAT))[lane];
  float4 b  = ((const float4*)bias)[lane];

  float4 v;
  v.x = a.x + hv.x * sn + b.x;
  v.y = a.y + hv.y * sn + b.y;
  v.z = a.z + hv.z * sn + b.z;
  v.w = a.w + hv.w * sn + b.w;

  // mean
  float s = v.x + v.y + v.z + v.w;
#pragma unroll
  for (int off = 16; off >= 1; off >>= 1) s += __shfl_xor(s, off, 32);
  float mu = s * (1.0f / (float)N_FEAT);

  // variance
  float dx = v.x - mu, dy = v.y - mu, dz = v.z - mu, dw = v.w - mu;
  float q = dx * dx + dy * dy + dz * dz + dw * dw;
#pragma unroll
  for (int off = 16; off >= 1; off >>= 1) q += __shfl_xor(q, off, 32);
  float rstd = rsqrtf(q * (1.0f / (float)N_FEAT) + LN_EPS);

  float4 g  = ((const float4*)gamma)[lane];
  float4 be = ((const float4*)beta)[lane];
  float4 y;
  y.x = fmaxf(dx * rstd * g.x + be.x, 0.0f);
  y.y = fmaxf(dy * rstd * g.y + be.y, 0.0f);
  y.z = fmaxf(dz * rstd * g.z + be.z, 0.0f);
  y.w = fmaxf(dw * rstd * g.w + be.w, 0.0f);

  ((float4*)(out + (size_t)node * N_FEAT))[lane] = y;
}

// ---------------------------------------------------------------------------
extern "C" void kernel_launch(void* const* d_in, const int* in_sizes, int n_in,
                              void* d_out, int out_size, void* d_ws, size_t ws_size,
                              hipStream_t stream) {
  const float* x   = (const float*)d_in[0];
  const int*   ei  = (const int*)d_in[1];
  const float* W1  = (const float*)d_in[2];
  const float* b1  = (const float*)d_in[3];
  const float* g1  = (const float*)d_in[4];
  const float* be1 = (const float*)d_in[5];
  const float* W2  = (const float*)d_in[6];
  const float* b2  = (const float*)d_in[7];
  const float* g2  = (const float*)d_in[8];
  const float* be2 = (const float*)d_in[9];

  const int n_nodes = in_sizes[0] / N_FEAT;
  const int n_edges = in_sizes[1] / 2;
  const int* src = ei;             // edge_index[0]
  const int* dst = ei + n_edges;   // edge_index[1]

  float* out = (float*)d_out;

  // workspace: dinv[N] | bufA[N*128] | bufB[N*128]
  char* ws = (char*)d_ws;
  size_t dinv_bytes = (((size_t)n_nodes * sizeof(float)) + 255) & ~(size_t)255;
  float* dinv = (float*)ws;
  float* bufA = (float*)(ws + dinv_bytes);
  float* bufB = bufA + (size_t)n_nodes * N_FEAT;

  const int BLK = 256;
  int feat_total = n_nodes * N_FEAT;
  int grid_feat  = (feat_total + BLK - 1) / BLK;
  int grid_edge  = (n_edges + BLK - 1) / BLK;
  int grid_node  = (n_nodes + BLK - 1) / BLK;
  int strips     = (n_nodes + 15) / 16;
  int grid_gemm  = (strips + 7) / 8;                  // 8 waves / block
  int grid_wedge = ((n_edges * 32) + BLK - 1) / BLK;  // wave per edge
  int grid_wnode = ((n_nodes * 32) + BLK - 1) / BLK;  // wave per node

  // degree / normalization
  init_kernel<<<grid_feat, BLK, 0, stream>>>(dinv, bufB, out, n_nodes);
  deg_accum_kernel<<<grid_edge, BLK, 0, stream>>>(dst, dinv, n_edges);
  deg_finalize_kernel<<<grid_node, BLK, 0, stream>>>(dinv, n_nodes);

  // ---- layer 1 ----
  gemm_wmma_f32_kernel<<<grid_gemm, BLK, 0, stream>>>(x, W1, bufA, n_nodes);
  scatter_kernel<<<grid_wedge, BLK, 0, stream>>>(src, dst, dinv, bufA, bufB, n_edges);
  combine_ln_relu_kernel<<<grid_wnode, BLK, 0, stream>>>(bufB, bufA, dinv, b1, g1, be1,
                                                         bufB, n_nodes);

  // ---- layer 2 ----  (agg accumulates directly into d_out, zeroed in init)
  gemm_wmma_f32_kernel<<<grid_gemm, BLK, 0, stream>>>(bufB, W2, bufA, n_nodes);
  scatter_kernel<<<grid_wedge, BLK, 0, stream>>>(src, dst, dinv, bufA, out, n_edges);
  combine_ln_relu_kernel<<<grid_wnode, BLK, 0, stream>>>(out, bufA, dinv, b2, g2, be2,
                                                         out, n_nodes);
}